// EGT_Layer_25623774888457
// MI455X (gfx1250) — compile-verified
//
#include <hip/hip_runtime.h>
#include <hip/hip_bf16.h>
#include <math.h>

// ---------------- constants ----------------
#define Bsz 16
#define Lsz 256
#define Dsz 512
#define EWsz 64
#define Hsz 32
#define DDsz 16
#define NIsz 1024
#define EIsz 128
#define ROWS_H (Bsz * Lsz)          // 4096
#define ROWS_E (Bsz * Lsz * Lsz)    // 1048576

typedef float v2f __attribute__((ext_vector_type(2)));
typedef float v8f __attribute__((ext_vector_type(8)));
typedef unsigned int u32x4 __attribute__((ext_vector_type(4)));
typedef int i32x4 __attribute__((ext_vector_type(4)));
typedef int i32x8 __attribute__((ext_vector_type(8)));

#if defined(__has_builtin)
#if __has_builtin(__builtin_amdgcn_tensor_load_to_lds) && \
    __has_builtin(__builtin_amdgcn_s_wait_tensorcnt)
#define HAVE_TDM 1
#endif
#endif
#ifndef HAVE_TDM
#define HAVE_TDM 0
#endif

static __device__ __forceinline__ v8f wmma4(v2f a, v2f b, v8f c) {
  // D = A(16x4,f32) * B(4x16,f32) + C(16x16,f32)  -> v_wmma_f32_16x16x4_f32
  return __builtin_amdgcn_wmma_f32_16x16x4_f32(false, a, false, b, (short)0, c,
                                               false, false);
}

// ---------------- LayerNorm over 512-wide rows ----------------
__global__ __launch_bounds__(256) void ln512_kernel(const float* __restrict__ x,
                                                    const float* __restrict__ g,
                                                    const float* __restrict__ b,
                                                    float* __restrict__ y) {
  int row = blockIdx.x;
  const float* xr = x + (size_t)row * Dsz;
  int t = threadIdx.x, wave = t >> 5, lane = t & 31;
  __shared__ float red[10];
  float v0 = xr[t], v1 = xr[t + 256];
  float s = v0 + v1;
  for (int off = 16; off >= 1; off >>= 1) s += __shfl_xor(s, off, 32);
  if (lane == 0) red[wave] = s;
  __syncthreads();
  if (t < 8) {
    float a = red[t];
    for (int off = 4; off >= 1; off >>= 1) a += __shfl_xor(a, off, 8);
    if (t == 0) red[8] = a;
  }
  __syncthreads();
  float mean = red[8] * (1.0f / Dsz);
  float d0 = v0 - mean, d1 = v1 - mean;
  float q = d0 * d0 + d1 * d1;
  for (int off = 16; off >= 1; off >>= 1) q += __shfl_xor(q, off, 32);
  __syncthreads();
  if (lane == 0) red[wave] = q;
  __syncthreads();
  if (t < 8) {
    float a = red[t];
    for (int off = 4; off >= 1; off >>= 1) a += __shfl_xor(a, off, 8);
    if (t == 0) red[9] = a;
  }
  __syncthreads();
  float rstd = rsqrtf(red[9] * (1.0f / Dsz) + 1e-5f);
  float* yr = y + (size_t)row * Dsz;
  yr[t]       = d0 * rstd * g[t] + b[t];
  yr[t + 256] = d1 * rstd * g[t + 256] + b[t + 256];
}

// ---------------- per-row mean/rstd of e (width 64) ----------------
__global__ __launch_bounds__(256) void moments64_kernel(const float* __restrict__ e,
                                                        float* __restrict__ mom) {
  int wave = threadIdx.x >> 5, lane = threadIdx.x & 31;
  size_t row = (size_t)blockIdx.x * 8 + wave;
  const float* xr = e + row * EWsz;
  float v0 = xr[lane], v1 = xr[lane + 32];
  float s = v0 + v1, q = v0 * v0 + v1 * v1;
  for (int off = 16; off >= 1; off >>= 1) {
    s += __shfl_xor(s, off, 32);
    q += __shfl_xor(q, off, 32);
  }
  if (lane == 0) {
    float m = s * (1.0f / EWsz);
    float var = q * (1.0f / EWsz) - m * m;
    mom[row * 2]     = m;
    mom[row * 2 + 1] = rsqrtf(var + 1e-5f);
  }
}

// ---------------- pack W_E | W_G side by side ----------------
__global__ void prep_weg_kernel(const float* __restrict__ WE, const float* __restrict__ bE,
                                const float* __restrict__ WG, const float* __restrict__ bG,
                                float* __restrict__ WEG, float* __restrict__ bEG) {
  int t = threadIdx.x;
  for (int i = t; i < EWsz * 64; i += 256) {
    int k = i >> 6, c = i & 63;
    WEG[i] = (c < 32) ? WE[k * 32 + c] : WG[k * 32 + (c - 32)];
  }
  if (t < 64) bEG[t] = (t < 32) ? bE[t] : bG[t - 32];
}

// ---------------- generic WMMA-f32 GEMM, 64x64 block tile ----------------
// EPI: 0 = bias   1 = bias + residual   2 = elu(bias)
template <bool LN_A, int EPI>
__global__ __launch_bounds__(256) void gemm_wmma_kernel(
    const float* __restrict__ A, int lda, const float* __restrict__ Bm, int ldb,
    const float* __restrict__ bias, const float* __restrict__ resid,
    const float* __restrict__ mom, const float* __restrict__ lng,
    const float* __restrict__ lnb, float* __restrict__ C, int ldc, int N, int K) {
  __shared__ float Ash[64][16];
  __shared__ float Bsh[16][64];
  int m0 = blockIdx.y * 64, n0 = blockIdx.x * 64;
  int t = threadIdx.x, wave = t >> 5, lane = t & 31;
  int hi = lane >> 4, lo = lane & 15;
  int t0 = wave * 2, t1 = wave * 2 + 1;  // two 16x16 tiles per wave (4x4 grid)
  int mr0 = (t0 >> 2) * 16, nc0 = (t0 & 3) * 16;
  int mr1 = (t1 >> 2) * 16, nc1 = (t1 & 3) * 16;
  v8f c0 = {0.f, 0.f, 0.f, 0.f, 0.f, 0.f, 0.f, 0.f};
  v8f c1 = {0.f, 0.f, 0.f, 0.f, 0.f, 0.f, 0.f, 0.f};

  for (int k0 = 0; k0 < K; k0 += 16) {
    __syncthreads();
    {  // A tile 64x16, one float4 per thread (optional fused LayerNorm)
      int row = t >> 2, cb = (t & 3) * 4;
      float4 av = *(const float4*)(A + (size_t)(m0 + row) * lda + k0 + cb);
      if (LN_A) {
        float m = mom[(size_t)(m0 + row) * 2];
        float r = mom[(size_t)(m0 + row) * 2 + 1];
        av.x = (av.x - m) * r * lng[k0 + cb + 0] + lnb[k0 + cb + 0];
        av.y = (av.y - m) * r * lng[k0 + cb + 1] + lnb[k0 + cb + 1];
        av.z = (av.z - m) * r * lng[k0 + cb + 2] + lnb[k0 + cb + 2];
        av.w = (av.w - m) * r * lng[k0 + cb + 3] + lnb[k0 + cb + 3];
      }
      *(float4*)&Ash[row][cb] = av;
      // prefetch next K-slab of A into cache (global_prefetch_b8)
      if (k0 + 16 < K)
        __builtin_prefetch(A + (size_t)(m0 + row) * lda + k0 + 16 + cb, 0, 0);
    }
    {  // B tile 16x64, one float4 per thread
      int row = t >> 4, cb = (t & 15) * 4;
      float4 bv;
      if (n0 + cb + 3 < N) {
        bv = *(const float4*)(Bm + (size_t)(k0 + row) * ldb + n0 + cb);
      } else {
        float tmp[4];
        for (int j = 0; j < 4; ++j) {
          int n = n0 + cb + j;
          tmp[j] = (n < N) ? Bm[(size_t)(k0 + row) * ldb + n] : 0.f;
        }
        bv = make_float4(tmp[0], tmp[1], tmp[2], tmp[3]);
      }
      *(float4*)&Bsh[row][cb] = bv;
      if (k0 + 16 < K)
        __builtin_prefetch(Bm + (size_t)(k0 + 16 + row) * ldb + n0 + cb, 0, 0);
    }
    __syncthreads();
    for (int ks = 0; ks < 16; ks += 4) {
      int ka = ks + hi * 2;
      v2f a0 = {Ash[mr0 + lo][ka], Ash[mr0 + lo][ka + 1]};
      v2f a1 = {Ash[mr1 + lo][ka], Ash[mr1 + lo][ka + 1]};
      v2f b0 = {Bsh[ka][nc0 + lo], Bsh[ka + 1][nc0 + lo]};
      v2f b1 = {Bsh[ka][nc1 + lo], Bsh[ka + 1][nc1 + lo]};
      c0 = wmma4(a0, b0, c0);
      c1 = wmma4(a1, b1, c1);
    }
  }
  // epilogue
  for (int tile = 0; tile < 2; ++tile) {
    v8f c = tile ? c1 : c0;
    int mr = tile ? mr1 : mr0, nc = tile ? nc1 : nc0;
    for (int r = 0; r < 8; ++r) {
      int mm = m0 + mr + r + hi * 8;
      int nn = n0 + nc + lo;
      if (nn < N) {
        float v = c[r] + bias[nn];
        if (EPI == 1) v += resid[(size_t)mm * ldc + nn];
        if (EPI == 2) v = v > 0.f ? v : expm1f(v);
        C[(size_t)mm * ldc + nn] = v;
      }
    }
  }
}

// ---------------- fused attention per (b, h, 32-row l-strip) ----------------
__global__ __launch_bounds__(256) void attn_kernel(const float* __restrict__ QKV,
                                                   const float* __restrict__ EG,
                                                   const float* __restrict__ mask,
                                                   float* __restrict__ Hh,
                                                   float* __restrict__ Vatt) {
  __shared__ float Ssh[32][256];    // scores / A_tild
  __shared__ float KVsh[256][16];   // K, then reused for V
  __shared__ float Qsh[32][16];
  __shared__ float Psh[2][4][256];  // AV partials (2 l-strips x 4 k-slices)
  int blk = blockIdx.x;
  int l0 = (blk & 7) * 32;
  int bh = blk >> 3;
  int h = bh & 31;
  int b = bh >> 5;
  int t = threadIdx.x, wave = t >> 5, lane = t & 31;
  int hi = lane >> 4, lo = lane & 15;
  const size_t qkvbase = (size_t)b * Lsz * (3 * Dsz);

  for (int i = t; i < 32 * 16; i += 256) {
    int r = i >> 4, d = i & 15;
    Qsh[r][d] = QKV[qkvbase + (size_t)(l0 + r) * 1536 + d * 32 + h];
  }
  for (int i = t; i < 256 * 16; i += 256) {
    int m = i >> 4, d = i & 15;
    KVsh[m][d] = QKV[qkvbase + (size_t)m * 1536 + 512 + d * 32 + h];
  }
  __syncthreads();

  // S = clip(Q K^T * DD^-0.5) + E ; also store H_hat for the edge path
  const float scale = 0.25f;  // DD^-0.5
  for (int i = 0; i < 4; ++i) {
    int tile = wave * 4 + i;
    int ls = tile >> 4;   // l-strip 0..1
    int ms = tile & 15;   // m-strip 0..15
    v8f c = {0.f, 0.f, 0.f, 0.f, 0.f, 0.f, 0.f, 0.f};
    for (int ks = 0; ks < 16; ks += 4) {
      int ka = ks + hi * 2;
      v2f a = {Qsh[ls * 16 + lo][ka], Qsh[ls * 16 + lo][ka + 1]};
      v2f bb = {KVsh[ms * 16 + lo][ka], KVsh[ms * 16 + lo][ka + 1]};
      c = wmma4(a, bb, c);
    }
    for (int r = 0; r < 8; ++r) {
      int lrow = ls * 16 + r + hi * 8;
      int m = ms * 16 + lo;
      int l = l0 + lrow;
      size_t eidx = (size_t)(b * Lsz + l) * Lsz + m;
      float v = c[r] * scale;
      v = fminf(fmaxf(v, -5.f), 5.f) + EG[eidx * 64 + h];
      Hh[eidx * 32 + h] = v;
      Ssh[lrow][m] = v;
    }
  }
  __syncthreads();

  // softmax over m (+mask) * sigmoid(G+mask); 8 threads per row
  {
    int r = t >> 3, sub = t & 7;
    int l = l0 + r;
    size_t rowbase = (size_t)(b * Lsz + l) * Lsz;
    float mx = -3.4e38f;
    for (int j = 0; j < 32; ++j) {
      int m = sub + j * 8;
      mx = fmaxf(mx, Ssh[r][m] + mask[rowbase + m]);
    }
    for (int off = 4; off >= 1; off >>= 1) mx = fmaxf(mx, __shfl_xor(mx, off, 8));
    float sum = 0.f;
    for (int j = 0; j < 32; ++j) {
      int m = sub + j * 8;
      sum += expf(Ssh[r][m] + mask[rowbase + m] - mx);
    }
    for (int off = 4; off >= 1; off >>= 1) sum += __shfl_xor(sum, off, 8);
    float inv = 1.f / sum;
    for (int j = 0; j < 32; ++j) {
      int m = sub + j * 8;
      float mk = mask[rowbase + m];
      float g = EG[(rowbase + m) * 64 + 32 + h];
      float gate = 1.f / (1.f + expf(-(g + mk)));
      Ssh[r][m] = expf(Ssh[r][m] + mk - mx) * inv * gate;
    }
  }
  __syncthreads();

  // load V into the K slot
  for (int i = t; i < 256 * 16; i += 256) {
    int m = i >> 4, d = i & 15;
    KVsh[m][d] = QKV[qkvbase + (size_t)m * 1536 + 1024 + d * 32 + h];
  }
  __syncthreads();

  // V_att = A_tild @ V   (M=32 l, N=16 d, K=256 m split over 4 wave-slices)
  {
    int ls = wave & 1, kslice = wave >> 1;
    v8f c = {0.f, 0.f, 0.f, 0.f, 0.f, 0.f, 0.f, 0.f};
    for (int k0 = kslice * 64; k0 < kslice * 64 + 64; k0 += 4) {
      int ka = k0 + hi * 2;
      v2f a = {Ssh[ls * 16 + lo][ka], Ssh[ls * 16 + lo][ka + 1]};
      v2f bb = {KVsh[ka][lo], KVsh[ka + 1][lo]};
      c = wmma4(a, bb, c);
    }
    for (int r = 0; r < 8; ++r) Psh[ls][kslice][(r + hi * 8) * 16 + lo] = c[r];
  }
  __syncthreads();
  for (int i = t; i < 512; i += 256) {
    int ls = i >> 8, e2 = i & 255;
    float s = Psh[ls][0][e2] + Psh[ls][1][e2] + Psh[ls][2][e2] + Psh[ls][3][e2];
    int l = l0 + ls * 16 + (e2 >> 4);
    int d = e2 & 15;
    Vatt[(size_t)(b * Lsz + l) * Dsz + d * 32 + h] = s;
  }
}

// ---------------- fused edge FFN (LN -> 64x128 ELU -> 128x64 + residual), in place
// Input tile (64 rows x 64 f32, contiguous 16KB) is staged into LDS via the
// Tensor Data Mover; wave 0 issues TENSOR_LOAD_TO_LDS and waits on TENSORcnt.
__global__ __launch_bounds__(256) void edge_ffn_kernel(float* __restrict__ eio,
                                                       const float* __restrict__ lng,
                                                       const float* __restrict__ lnb,
                                                       const float* __restrict__ W1,
                                                       const float* __restrict__ b1,
                                                       const float* __restrict__ W2,
                                                       const float* __restrict__ b2) {
  __shared__ float Ash[64][64];    // LN'ed input tile
  __shared__ float Tsh[64][128];   // hidden activations
  int t = threadIdx.x, wave = t >> 5, lane = t & 31;
  int hi = lane >> 4, lo = lane & 15;
  size_t row0 = (size_t)blockIdx.x * 64;

#if HAVE_TDM
  if (wave == 0) {
    // D# group0: count=1 | lds_addr | global_addr[56:0] | type=2
    unsigned lds_off = (unsigned)(unsigned long long)(&Ash[0][0]);  // low 32 bits = LDS byte offset
    unsigned long long ga = (unsigned long long)(eio + row0 * EWsz);
    u32x4 g0;
    g0[0] = 1u;                                   // count=1, is_restore=0, gather off
    g0[1] = lds_off;                              // lds_addr
    g0[2] = (unsigned)(ga & 0xFFFFFFFFull);       // global_addr[31:0]
    g0[3] = (unsigned)((ga >> 32) & 0x1FFFFFFull) // global_addr[56:32]
            | (2u << 30);                         // type=2 ("image")
    // D# group1: data_size=4B, tensor 64 x ROWS_E, tile 64x64, dim0 stride 64
    i32x8 g1;
    g1[0] = (2 << 16);                            // data_size=2 (4 bytes)
    g1[1] = (int)(EWsz << 16);                    // tensor_dim0[15:0]=64 at bits 63:48
    g1[2] = 0;                                    // tensor_dim0 hi=0, tensor_dim1 lo16=0
    g1[3] = (int)((ROWS_E >> 16) & 0xFFFF)        // tensor_dim1[31:16] (=16)
            | (EWsz << 16);                       // tile_dim0=64
    g1[4] = 64;                                   // tile_dim1=64, tile_dim2=0
    g1[5] = EWsz;                                 // tensor_dim0_stride=64 (low 32)
    g1[6] = 0;
    g1[7] = 0;
    i32x4 gz4 = {0, 0, 0, 0};                     // groups 2/3 unused (2-D tensor)
    i32x8 gz8 = {0, 0, 0, 0, 0, 0, 0, 0};
    __builtin_amdgcn_tensor_load_to_lds(g0, g1, gz4, gz4, gz8, 0);
    __builtin_amdgcn_s_wait_tensorcnt(0);
  }
#else
  for (int i = t; i < 64 * 64 / 4; i += 256) {
    int idx = i * 4, r = idx >> 6, c = idx & 63;
    *(float4*)&Ash[r][c] = *(const float4*)&eio[(row0 + r) * EWsz + c];
  }
#endif
  __syncthreads();
  {  // per-row LN (4 threads/row)
    int r = t >> 2, sub = t & 3;
    float s = 0.f, q = 0.f;
    for (int j = 0; j < 16; ++j) {
      float v = Ash[r][sub * 16 + j];
      s += v; q += v * v;
    }
    for (int off = 2; off >= 1; off >>= 1) {
      s += __shfl_xor(s, off, 4);
      q += __shfl_xor(q, off, 4);
    }
    float m = s * (1.0f / 64.f);
    float rstd = rsqrtf(q * (1.0f / 64.f) - m * m + 1e-5f);
    for (int j = 0; j < 16; ++j) {
      int c = sub * 16 + j;
      Ash[r][c] = (Ash[r][c] - m) * rstd * lng[c] + lnb[c];
    }
  }
  __syncthreads();
  // GEMM1: [64x64] @ [64x128] -> elu -> Tsh (B fragments straight from global)
  for (int i = 0; i < 4; ++i) {
    int tile = wave * 4 + i;
    int mr = (tile >> 3) * 16, nc = (tile & 7) * 16;
    v8f c = {0.f, 0.f, 0.f, 0.f, 0.f, 0.f, 0.f, 0.f};
    for (int k0 = 0; k0 < 64; k0 += 4) {
      int ka = k0 + hi * 2;
      v2f a = {Ash[mr + lo][ka], Ash[mr + lo][ka + 1]};
      v2f bb = {W1[(size_t)ka * EIsz + nc + lo], W1[(size_t)(ka + 1) * EIsz + nc + lo]};
      c = wmma4(a, bb, c);
    }
    for (int r = 0; r < 8; ++r) {
      int mm = mr + r + hi * 8, nn = nc + lo;
      float v = c[r] + b1[nn];
      Tsh[mm][nn] = v > 0.f ? v : expm1f(v);
    }
  }
  __syncthreads();
  // GEMM2: [64x128] @ [128x64] + bias + residual (residual re-read from global)
  for (int i = 0; i < 2; ++i) {
    int tile = wave * 2 + i;
    int mr = (tile >> 2) * 16, nc = (tile & 3) * 16;
    v8f c = {0.f, 0.f, 0.f, 0.f, 0.f, 0.f, 0.f, 0.f};
    for (int k0 = 0; k0 < 128; k0 += 4) {
      int ka = k0 + hi * 2;
      v2f a = {Tsh[mr + lo][ka], Tsh[mr + lo][ka + 1]};
      v2f bb = {W2[(size_t)ka * EWsz + nc + lo], W2[(size_t)(ka + 1) * EWsz + nc + lo]};
      c = wmma4(a, bb, c);
    }
    for (int r = 0; r < 8; ++r) {
      int mm = mr + r + hi * 8, nn = nc + lo;
      size_t gi = (row0 + mm) * EWsz + nn;
      eio[gi] = c[r] + b2[nn] + eio[gi];
    }
  }
}

// ---------------- host launch ----------------
extern "C" void kernel_launch(void* const* d_in, const int* in_sizes, int n_in,
                              void* d_out, int out_size, void* d_ws, size_t ws_size,
                              hipStream_t stream) {
  (void)in_sizes; (void)n_in; (void)out_size; (void)ws_size;
  const float* h        = (const float*)d_in[0];
  const float* e        = (const float*)d_in[1];
  const float* mask     = (const float*)d_in[2];
  const float* ln_h_g   = (const float*)d_in[3];
  const float* ln_h_b   = (const float*)d_in[4];
  const float* ln_e_g   = (const float*)d_in[5];
  const float* ln_e_b   = (const float*)d_in[6];
  const float* W_QKV    = (const float*)d_in[7];
  const float* b_QKV    = (const float*)d_in[8];
  const float* W_E      = (const float*)d_in[9];
  const float* b_E      = (const float*)d_in[10];
  const float* W_G      = (const float*)d_in[11];
  const float* b_G      = (const float*)d_in[12];
  const float* W_Oh     = (const float*)d_in[13];
  const float* b_Oh     = (const float*)d_in[14];
  const float* f_ln_h_g = (const float*)d_in[15];
  const float* f_ln_h_b = (const float*)d_in[16];
  const float* W_h1     = (const float*)d_in[17];
  const float* b_h1     = (const float*)d_in[18];
  const float* W_h2     = (const float*)d_in[19];
  const float* b_h2     = (const float*)d_in[20];
  const float* W_Oe     = (const float*)d_in[21];
  const float* b_Oe     = (const float*)d_in[22];
  const float* f_ln_e_g = (const float*)d_in[23];
  const float* f_ln_e_b = (const float*)d_in[24];
  const float* W_e1     = (const float*)d_in[25];
  const float* b_e1     = (const float*)d_in[26];
  const float* W_e2     = (const float*)d_in[27];
  const float* b_e2     = (const float*)d_in[28];

  float* ws = (float*)d_ws;
  float* h_ln  = ws;                       // 4096*512
  float* h_ln2 = h_ln  + (size_t)ROWS_H * Dsz;
  float* QKV   = h_ln2 + (size_t)ROWS_H * Dsz;          // 4096*1536
  float* Vatt  = QKV   + (size_t)ROWS_H * 3 * Dsz;      // 4096*512
  float* t_h   = Vatt  + (size_t)ROWS_H * Dsz;          // 4096*1024
  float* mom_e = t_h   + (size_t)ROWS_H * NIsz;         // 1M*2
  float* W_EG  = mom_e + (size_t)ROWS_E * 2;            // 64*64
  float* b_EG  = W_EG  + 64 * 64;                       // 64
  float* Hh    = b_EG  + 64;                            // 1M*32
  float* EG    = Hh    + (size_t)ROWS_E * Hsz;          // 1M*64

  float* out_h = (float*)d_out;                          // 4096*512
  float* out_e = out_h + (size_t)ROWS_H * Dsz;           // 1M*64

  // ---- node: LN(h), QKV projection ----
  ln512_kernel<<<ROWS_H, 256, 0, stream>>>(h, ln_h_g, ln_h_b, h_ln);
  gemm_wmma_kernel<false, 0><<<dim3(1536 / 64, ROWS_H / 64), 256, 0, stream>>>(
      h_ln, Dsz, W_QKV, 1536, b_QKV, nullptr, nullptr, nullptr, nullptr,
      QKV, 1536, 1536, Dsz);

  // ---- edge: moments, packed E|G projection with fused LN ----
  moments64_kernel<<<ROWS_E / 8, 256, 0, stream>>>(e, mom_e);
  prep_weg_kernel<<<1, 256, 0, stream>>>(W_E, b_E, W_G, b_G, W_EG, b_EG);
  gemm_wmma_kernel<true, 0><<<dim3(1, ROWS_E / 64), 256, 0, stream>>>(
      e, EWsz, W_EG, 64, b_EG, nullptr, mom_e, ln_e_g, ln_e_b,
      EG, 64, 64, EWsz);

  // ---- attention (writes H_hat and V_att) ----
  attn_kernel<<<Bsz * Hsz * (Lsz / 32), 256, 0, stream>>>(QKV, EG, mask, Hh, Vatt);

  // ---- node output projection + residual, FFN ----
  gemm_wmma_kernel<false, 1><<<dim3(Dsz / 64, ROWS_H / 64), 256, 0, stream>>>(
      Vatt, Dsz, W_Oh, Dsz, b_Oh, h, nullptr, nullptr, nullptr,
      out_h, Dsz, Dsz, Dsz);
  ln512_kernel<<<ROWS_H, 256, 0, stream>>>(out_h, f_ln_h_g, f_ln_h_b, h_ln2);
  gemm_wmma_kernel<false, 2><<<dim3(NIsz / 64, ROWS_H / 64), 256, 0, stream>>>(
      h_ln2, Dsz, W_h1, NIsz, b_h1, nullptr, nullptr, nullptr, nullptr,
      t_h, NIsz, NIsz, Dsz);
  gemm_wmma_kernel<false, 1><<<dim3(Dsz / 64, ROWS_H / 64), 256, 0, stream>>>(
      t_h, NIsz, W_h2, Dsz, b_h2, out_h, nullptr, nullptr, nullptr,
      out_h, Dsz, Dsz, NIsz);

  // ---- edge output projection + residual, then fused edge FFN in place ----
  gemm_wmma_kernel<false, 1><<<dim3(1, ROWS_E / 64), 256, 0, stream>>>(
      Hh, Hsz, W_Oe, EWsz, b_Oe, e, nullptr, nullptr, nullptr,
      out_e, EWsz, EWsz, Hsz);
  edge_ffn_kernel<<<ROWS_E / 64, 256, 0, stream>>>(out_e, f_ln_e_g, f_ln_e_b,
                                                   W_e1, b_e1, W_e2, b_e2);
}